// FrequencyDecomposition_24567212933361
// MI455X (gfx1250) — compile-verified
//
#include <hip/hip_runtime.h>

// ---------------------------------------------------------------------------
// Frequency decomposition (8x8 block DCT -> band mask -> IDCT) for gfx1250.
//
// Strategy (see analysis): 16x16 tiles = 2x2 grid of 8x8 DCT blocks, with
// Dblk = blockdiag(D8, D8). All matmuls are V_WMMA_F32_16X16X4_F32 chains
// (K=16 -> 4 wmma). Transposes between separable passes are realized by
// feeding the C/D-layout result as the next WMMA's A operand, which only
// needs half-wave swaps (ds_swizzle xor 16) + lane selects.
// mid band = X - low - high (masks partition), saving 2 of 8 transform P2s.
// ---------------------------------------------------------------------------

typedef __attribute__((ext_vector_type(2))) float v2f;
typedef __attribute__((ext_vector_type(8))) float v8f;

// ds_swizzle BITMASK_PERM: offset[14:10]=and=0x1F, [9:5]=or=0, [4:0]=xor=0x10
// -> src lane = lane ^ 16 (swap wave halves)
__device__ __forceinline__ float swap_half(float x) {
  return __builtin_bit_cast(
      float, __builtin_amdgcn_ds_swizzle(__builtin_bit_cast(int, x), 0x7C10));
}

// Orthonormal DCT-II entry D[k][i], n=8. sqrt(2/8)=0.5, D[0][*]=1/sqrt(8).
__device__ __forceinline__ float dct_entry(int k, int i) {
  if (k == 0) return 0.35355339059327379f;
  return 0.5f * cosf(3.14159265358979323846f * (float)(k * (2 * i + 1)) / 16.0f);
}

// P2: r = S^T * E   (S given in WMMA C-layout, E given per-lane in B-layout)
// C-layout of S: vgpr j, lanes 0-15 -> row j, lanes 16-31 -> row j+8, col=lane%16.
// Using S as the A operand realizes S^T; the only data movement is:
//   a0={s0.lo,s2.lo} a1={s1.lo,s3.lo} a2={s4.lo,s6.lo} a3={s5.lo,s7.lo}
//   a4={s0.hi,s2.hi} a5={s1.hi,s3.hi} a6={s4.hi,s6.hi} a7={s5.hi,s7.hi}
__device__ __forceinline__ void p2(const float s[8], const float e[8], bool hi,
                                   float r[8]) {
  float a0, a1, a2, a3, a4, a5, a6, a7;
  {
    float w0 = swap_half(s[0]), w2 = swap_half(s[2]);
    a0 = hi ? w2 : s[0];
    a4 = hi ? s[2] : w0;
  }
  {
    float w1 = swap_half(s[1]), w3 = swap_half(s[3]);
    a1 = hi ? w3 : s[1];
    a5 = hi ? s[3] : w1;
  }
  {
    float w4 = swap_half(s[4]), w6 = swap_half(s[6]);
    a2 = hi ? w6 : s[4];
    a6 = hi ? s[6] : w4;
  }
  {
    float w5 = swap_half(s[5]), w7 = swap_half(s[7]);
    a3 = hi ? w7 : s[5];
    a7 = hi ? s[7] : w5;
  }
  v8f acc = {0.f, 0.f, 0.f, 0.f, 0.f, 0.f, 0.f, 0.f};
  v2f av, bv;
  av.x = a0; av.y = a1; bv.x = e[0]; bv.y = e[1];
  acc = __builtin_amdgcn_wmma_f32_16x16x4_f32(false, av, false, bv, (short)0,
                                              acc, false, false);
  av.x = a2; av.y = a3; bv.x = e[2]; bv.y = e[3];
  acc = __builtin_amdgcn_wmma_f32_16x16x4_f32(false, av, false, bv, (short)0,
                                              acc, false, false);
  av.x = a4; av.y = a5; bv.x = e[4]; bv.y = e[5];
  acc = __builtin_amdgcn_wmma_f32_16x16x4_f32(false, av, false, bv, (short)0,
                                              acc, false, false);
  av.x = a6; av.y = a7; bv.x = e[6]; bv.y = e[7];
  acc = __builtin_amdgcn_wmma_f32_16x16x4_f32(false, av, false, bv, (short)0,
                                              acc, false, false);
#pragma unroll
  for (int j = 0; j < 8; ++j) r[j] = acc[j];
}

// Zigzag band bitmasks packed per column l (lane&7): bit j = mask[row j][l].
// low: z<16, high: z>=48 (JPEG zigzag, matches reference generator).
#define LOWM_PACKED 0x0000010103070F1FULL
#define HIGHM_PACKED 0xF8F0E0C080800000ULL

__global__ __launch_bounds__(256) void freq_decomp_kernel(
    const float* __restrict__ x, float* __restrict__ out, int numTiles,
    int numWaves) {
  const int lane = threadIdx.x & 31;
  const int wave = blockIdx.x * (blockDim.x >> 5) + (threadIdx.x >> 5);
  const int n = lane & 15;
  const bool hi = lane >= 16;

  // Constant B operands, per-lane. B-layout: vgpr j, lane L holds
  // E[k][n] with k = 4*(j>>1) + (j&1) + 2*(L>=16), n = L%16.
  // eF = Dblk^T (forward), eI = Dblk (inverse); Dblk = blockdiag(D8,D8).
  float eF[8], eI[8];
#pragma unroll
  for (int j = 0; j < 8; ++j) {
    int k = 4 * (j >> 1) + (j & 1) + (hi ? 2 : 0);
    bool diag = (k >> 3) == (n >> 3);
    int kr = k & 7, nc = n & 7;
    eI[j] = diag ? dct_entry(kr, nc) : 0.0f;  // Dblk[k][n]
    eF[j] = diag ? dct_entry(nc, kr) : 0.0f;  // Dblk^T[k][n]
  }
  const unsigned lowbits = (unsigned)((LOWM_PACKED >> (8 * (lane & 7))) & 0xFF);
  const unsigned highbits =
      (unsigned)((HIGHM_PACKED >> (8 * (lane & 7))) & 0xFF);

  const long long bandStride = (long long)numTiles * 256;  // B*C*H*W elems
  const int laneRow = (hi ? 8 : 0);

  for (int t = wave; t < numTiles; t += numWaves) {
    // 512x512 image = 32x32 tiles of 16x16
    const int img = t >> 10;
    const int tr = (t >> 5) & 31;
    const int tc = t & 31;
    const long long base =
        ((long long)img << 18) + (long long)(tr * 16 + laneRow) * 512 +
        (long long)(tc * 16 + n);

    // Load tile straight into C-layout: s[j] = X[row j + 8*hi][n]
    float s[8];
#pragma unroll
    for (int j = 0; j < 8; ++j) s[j] = x[base + (long long)j * 512];

    // Forward: coef = Dblk * X * Dblk^T  via two P2 with E = Dblk^T
    float t1[8], coef[8];
    p2(s, eF, hi, t1);
    p2(t1, eF, hi, coef);

    // Low band: rec_low = Dblk^T * (Mlow o coef) * Dblk
    float g[8], recl[8], rech[8];
#pragma unroll
    for (int j = 0; j < 8; ++j)
      g[j] = ((lowbits >> j) & 1u) ? coef[j] : 0.0f;
    p2(g, eI, hi, t1);
    p2(t1, eI, hi, recl);

    // High band
#pragma unroll
    for (int j = 0; j < 8; ++j)
      g[j] = ((highbits >> j) & 1u) ? coef[j] : 0.0f;
    p2(g, eI, hi, t1);
    p2(t1, eI, hi, rech);

    // Store: band0=low, band1=mid=X-low-high (masks partition), band2=high
#pragma unroll
    for (int j = 0; j < 8; ++j) {
      const long long o = base + (long long)j * 512;
      const float lo = recl[j], hv = rech[j];
      out[o] = lo;
      out[bandStride + o] = s[j] - lo - hv;
      out[2 * bandStride + o] = hv;
    }
  }
}

extern "C" void kernel_launch(void* const* d_in, const int* in_sizes, int n_in,
                              void* d_out, int out_size, void* d_ws,
                              size_t ws_size, hipStream_t stream) {
  const float* x = (const float*)d_in[0];
  float* out = (float*)d_out;
  const int numTiles = in_sizes[0] / 256;  // 16x16 pixels per tile
  if (numTiles <= 0) return;
  const int blocks = 4096;                  // persistent-ish, 8 waves/block
  const int numWaves = blocks * 8;
  freq_decomp_kernel<<<blocks, 256, 0, stream>>>(x, out, numTiles, numWaves);
}